// MultiModalHGAT_6949257085553
// MI455X (gfx1250) — compile-verified
//
#include <hip/hip_runtime.h>
#include <hip/hip_bf16.h>
#include <stddef.h>

// ---------------------------------------------------------------------------
// MultiModalHGAT forward for gfx1250 (MI455X, wave32, WMMA).
// All dense GEMMs run through a V_WMMA_F32_16X16X4_F32 tile kernel.
// ---------------------------------------------------------------------------

typedef __attribute__((ext_vector_type(2))) float v2f;
typedef __attribute__((ext_vector_type(8))) float v8f;

#define N_P  512
#define N_B  4096
#define E_PP 8192
#define E_BP 16384
#define E_BB 32768
#define D_P  34
#define D_B  16
#define D_E  16
#define HID  128
#define CHUNK 16

__device__ __forceinline__ float leaky(float x, float s) { return x > 0.f ? x : s * x; }

// monotonic float<->uint mapping for atomicMax on floats
__device__ __forceinline__ unsigned f2ord(float f) {
  unsigned u = __float_as_uint(f);
  return (u & 0x80000000u) ? ~u : (u | 0x80000000u);
}
__device__ __forceinline__ float ord2f(unsigned k) {
  unsigned u = (k & 0x80000000u) ? (k & 0x7fffffffu) : ~k;
  return __uint_as_float(u);
}
// host-side encode(-inf) = ~0xFF800000 = 0x007FFFFF
#define ORD_NEG_INF 0x007FFFFFu

// ---------------------------------------------------------------------------
// WMMA fp32 GEMM:  C[M,N] = act( A[M,K] @ W[N,K]^T + bias[N] )
// One wave (32 threads) per 16x16 output tile; K stepped by 4 via
// V_WMMA_F32_16X16X4_F32.
//   A 16x4 layout: lanes 0-15 -> (M=lane, K=k,k+1); lanes 16-31 -> (M, K=k+2,k+3)
//   B 4x16 layout: symmetric, N = lane&15
//   C/D:           VGPR r -> M = r (+8 for upper half), N = lane&15
//
// Out-of-range tile rows/cols: row pointers are clamped to row 0 (valid
// memory) and loaded UNCONDITIONALLY. Since D[m][n] depends only on A row m
// and B col n, the resulting garbage lands exclusively in C rows/cols that
// the store loop never writes -> no predication in the hot loop, no EXEC
// save/restore, b64 (float2) loads.
// ---------------------------------------------------------------------------
__global__ void k_gemm_wmma(const float* __restrict__ A, const float* __restrict__ W,
                            const float* __restrict__ bias, float* __restrict__ C,
                            int M, int N, int K, int ldc, int coff, int act)
{
  const int lane = threadIdx.x;
  const int half = lane >> 4;
  const int l    = lane & 15;
  const int m0 = blockIdx.y << 4;
  const int n0 = blockIdx.x << 4;
  const int am = m0 + l;
  const int bn = n0 + l;
  const float* __restrict__ Arow = A + (size_t)(am < M ? am : 0) * K;
  const float* __restrict__ Wrow = W + (size_t)(bn < N ? bn : 0) * K;

  v8f acc = {0.f, 0.f, 0.f, 0.f, 0.f, 0.f, 0.f, 0.f};
  const int Kq = K >> 2;              // number of full K=4 steps
  if ((K & 1) == 0) {
    // even K (all call sites): rows are 8-byte aligned -> b64 loads
    const float2* __restrict__ A2 = (const float2*)Arow + half;
    const float2* __restrict__ W2 = (const float2*)Wrow + half;
#pragma unroll 4
    for (int q = 0; q < Kq; ++q) {
      float2 a2 = A2[q << 1];
      float2 b2 = W2[q << 1];
      v2f av; av.x = a2.x; av.y = a2.y;
      v2f bv; bv.x = b2.x; bv.y = b2.y;
      acc = __builtin_amdgcn_wmma_f32_16x16x4_f32(false, av, false, bv,
                                                  (short)0, acc, false, false);
    }
  } else {
    const int koff = half << 1;
    for (int q = 0; q < Kq; ++q) {
      int k = (q << 2) + koff;
      v2f av; av.x = Arow[k]; av.y = Arow[k + 1];
      v2f bv; bv.x = Wrow[k]; bv.y = Wrow[k + 1];
      acc = __builtin_amdgcn_wmma_f32_16x16x4_f32(false, av, false, bv,
                                                  (short)0, acc, false, false);
    }
  }
  int k = Kq << 2;
  if (k < K) {  // K % 4 != 0 tail (e.g. K = 34); masks are uniform per half-wave
    v2f av = {0.f, 0.f}, bv = {0.f, 0.f};
    int k0 = k + (half << 1);
    if (k0 < K)     { av.x = Arow[k0];     bv.x = Wrow[k0];     }
    if (k0 + 1 < K) { av.y = Arow[k0 + 1]; bv.y = Wrow[k0 + 1]; }
    acc = __builtin_amdgcn_wmma_f32_16x16x4_f32(false, av, false, bv,
                                                (short)0, acc, false, false);
  }

  const int n = n0 + l;
  if (n >= N) return;
  const float bval = bias ? bias[n] : 0.f;
#pragma unroll
  for (int r = 0; r < 8; ++r) {
    int m = m0 + r + (half << 3);
    if (m < M) {
      float v = acc[r] + bval;
      if (act == 1) v = leaky(v, 0.01f);
      C[(size_t)m * ldc + coff + n] = v;
    }
  }
}

// ---------------------------------------------------------------------------
// Patch embed: conv(1->8,s2,SAME) -> lrelu -> conv(8->16,s2,SAME) -> lrelu
//              -> GAP -> linear(16->64)
// ---------------------------------------------------------------------------
__global__ void k_conv1(const float* __restrict__ x, const float* __restrict__ w,
                        const float* __restrict__ b, float* __restrict__ y, int nimg)
{
  const int OS = 25, IS = 50;
  int t = blockIdx.x * blockDim.x + threadIdx.x;
  int total = nimg * 8 * OS * OS * OS;
  if (t >= total) return;
  int xx = t % OS; int tt = t / OS;
  int yy = tt % OS; tt /= OS;
  int zz = tt % OS; tt /= OS;
  int c  = tt % 8;  int n = tt / 8;
  const float* __restrict__ in = x + (size_t)n * (IS * IS * IS);
  const float* __restrict__ wk = w + c * 27;
  float acc = b[c];
  // SAME, in=50,k=3,s=2 -> pad_lo=0, pad_hi=1
  for (int kd = 0; kd < 3; ++kd) {
    int iz = zz * 2 + kd; if (iz >= IS) continue;
    for (int kh = 0; kh < 3; ++kh) {
      int iy = yy * 2 + kh; if (iy >= IS) continue;
      for (int kw = 0; kw < 3; ++kw) {
        int ix = xx * 2 + kw; if (ix >= IS) continue;
        acc += in[(iz * IS + iy) * IS + ix] * wk[(kd * 3 + kh) * 3 + kw];
      }
    }
  }
  y[t] = leaky(acc, 0.01f);
}

__global__ void k_conv2(const float* __restrict__ x, const float* __restrict__ w,
                        const float* __restrict__ b, float* __restrict__ y, int nimg)
{
  const int OS = 13, IS = 25;
  int t = blockIdx.x * blockDim.x + threadIdx.x;
  int total = nimg * 16 * OS * OS * OS;
  if (t >= total) return;
  int xx = t % OS; int tt = t / OS;
  int yy = tt % OS; tt /= OS;
  int zz = tt % OS; tt /= OS;
  int c  = tt % 16; int n = tt / 16;
  float acc = b[c];
  // SAME, in=25,k=3,s=2 -> pad_lo=1, pad_hi=1
  for (int ci = 0; ci < 8; ++ci) {
    const float* __restrict__ in = x + (size_t)(n * 8 + ci) * (IS * IS * IS);
    const float* __restrict__ wk = w + (size_t)(c * 8 + ci) * 27;
    for (int kd = 0; kd < 3; ++kd) {
      int iz = zz * 2 + kd - 1; if ((unsigned)iz >= (unsigned)IS) continue;
      for (int kh = 0; kh < 3; ++kh) {
        int iy = yy * 2 + kh - 1; if ((unsigned)iy >= (unsigned)IS) continue;
        for (int kw = 0; kw < 3; ++kw) {
          int ix = xx * 2 + kw - 1; if ((unsigned)ix >= (unsigned)IS) continue;
          acc += in[(iz * IS + iy) * IS + ix] * wk[(kd * 3 + kh) * 3 + kw];
        }
      }
    }
  }
  y[t] = leaky(acc, 0.01f);
}

__global__ void k_gap(const float* __restrict__ x, float* __restrict__ pooled, int nimg)
{
  int t = blockIdx.x * blockDim.x + threadIdx.x;
  if (t >= nimg * 16) return;
  const int V = 13 * 13 * 13;
  const float* __restrict__ p = x + (size_t)t * V;
  float s = 0.f;
  for (int i = 0; i < V; ++i) s += p[i];
  pooled[t] = s * (1.f / (float)V);
}

__global__ void k_patch_linear(const float* __restrict__ pooled, const float* __restrict__ Wl,
                               const float* __restrict__ bl, float* __restrict__ hp)
{
  int t = blockIdx.x * blockDim.x + threadIdx.x;
  if (t >= N_P * 64) return;
  int n = t / 64, j = t % 64;
  float s = bl[j];
  for (int c = 0; c < 16; ++c) s += pooled[n * 16 + c] * Wl[j * 16 + c];
  hp[(size_t)n * HID + 64 + j] = s;  // concat: xp occupies cols 64..127
}

// ---------------------------------------------------------------------------
// GAT helper kernels
// ---------------------------------------------------------------------------
// w_att[h,k] = sum_f ae[h,f] * We[(h*F+f), k]     (folds (he*ae).sum(-1))
__global__ void k_watt(const float* __restrict__ ae, const float* __restrict__ We,
                       float* __restrict__ w_att, int H, int F)
{
  int t = blockIdx.x * blockDim.x + threadIdx.x;
  if (t >= H * HID) return;
  int h = t / HID, k = t % HID;
  float acc = 0.f;
  for (int f = 0; f < F; ++f)
    acc += ae[h * F + f] * We[(size_t)(h * F + f) * HID + k];
  w_att[t] = acc;
}

// esum[e,h] = efeat[e,:] . w_att[h,:]
__global__ void k_esum(const float* __restrict__ efeat, const float* __restrict__ w_att,
                       float* __restrict__ esum, int E, int H)
{
  int t = blockIdx.x * blockDim.x + threadIdx.x;
  if (t >= E * H) return;
  int e = t / H, h = t - e * H;
  const float* __restrict__ row = efeat + (size_t)e * HID;
  const float* __restrict__ wa  = w_att + h * HID;
  float acc = 0.f;
  for (int k = 0; k < HID; ++k) acc += row[k] * wa[k];
  esum[t] = acc;
}

// out[n,h] = sum_f hl[n, h*F+f] * attv[h*F+f]
__global__ void k_nodesum(const float* __restrict__ hl, const float* __restrict__ attv,
                          float* __restrict__ out, int Nn, int H, int F)
{
  int t = blockIdx.x * blockDim.x + threadIdx.x;
  if (t >= Nn * H) return;
  int n = t / H, h = t - n * H;
  const float* __restrict__ row = hl + (size_t)n * (H * F) + h * F;
  const float* __restrict__ av  = attv + h * F;
  float acc = 0.f;
  for (int f = 0; f < F; ++f) acc += row[f] * av[f];
  out[t] = acc;
}

__global__ void k_colmean(const float* __restrict__ efeat, float* __restrict__ mean, int E)
{
  int k = threadIdx.x;  // blockDim == HID
  float s = 0.f;
  for (int e = 0; e < E; ++e) s += efeat[(size_t)e * HID + k];
  mean[k] = s / (float)E;
}

__global__ void k_loop_att(const float* __restrict__ mean, const float* __restrict__ w_att,
                           float* __restrict__ loop_att, int H)
{
  int h = threadIdx.x;
  if (h >= H) return;
  float acc = 0.f;
  for (int k = 0; k < HID; ++k) acc += mean[k] * w_att[h * HID + k];
  loop_att[h] = acc;
}

__global__ void k_fill_u32(unsigned* __restrict__ p, unsigned v, int n)
{
  int t = blockIdx.x * blockDim.x + threadIdx.x;
  if (t < n) p[t] = v;
}

// attention logit + leaky(0.2) + segment max (into m_ord)
__global__ void k_attn(const float* __restrict__ ssum, const float* __restrict__ dsum,
                       const float* __restrict__ esum, const float* __restrict__ loop_att,
                       const int* __restrict__ src, const int* __restrict__ dst,
                       int E, int n_loop, int H,
                       float* __restrict__ a, unsigned* __restrict__ m_ord)
{
  int t = blockIdx.x * blockDim.x + threadIdx.x;
  int Etot = E + n_loop;
  if (t >= Etot * H) return;
  int e = t / H, h = t - e * H;
  int s, d; float base;
  if (e < E) { s = src[e]; d = dst[e]; base = esum[e * H + h]; }
  else       { s = d = e - E;          base = loop_att[h];     }
  float v = ssum[s * H + h] + dsum[d * H + h] + base;
  v = leaky(v, 0.2f);
  a[t] = v;
  atomicMax(&m_ord[d * H + h], f2ord(v));
}

__global__ void k_m_decode(const unsigned* __restrict__ m_ord, float* __restrict__ m_f, int n)
{
  int t = blockIdx.x * blockDim.x + threadIdx.x;
  if (t >= n) return;
  float v = ord2f(m_ord[t]);
  m_f[t] = isfinite(v) ? v : 0.f;   // jnp.where(isfinite(m), m, 0)
}

// ea = exp(a - m[dst]) in-place; z[dst] += ea
__global__ void k_ea(float* __restrict__ a, const float* __restrict__ m_f,
                     const int* __restrict__ dst, int E, int n_loop, int H,
                     float* __restrict__ z)
{
  int t = blockIdx.x * blockDim.x + threadIdx.x;
  int Etot = E + n_loop;
  if (t >= Etot * H) return;
  int e = t / H, h = t - e * H;
  int d = (e < E) ? dst[e] : (e - E);
  float v = expf(a[t] - m_f[d * H + h]);
  a[t] = v;
  atomicAdd(&z[d * H + h], v);
}

// agg[dst, :] += (ea/(z+eps)) * hs_l[src, :]   (one block per edge, HF threads)
__global__ void k_message(const float* __restrict__ ea, const float* __restrict__ z,
                          const float* __restrict__ hs_l,
                          const int* __restrict__ src, const int* __restrict__ dst,
                          int E, int H, int F, float* __restrict__ agg)
{
  int e = blockIdx.x;
  int f = threadIdx.x;          // f < H*F
  int h = f / F;
  int s, d;
  if (e < E) { s = src[e]; d = dst[e]; }
  else       { s = d = e - E; }
  float alpha = ea[e * H + h] / (z[d * H + h] + 1e-16f);
  atomicAdd(&agg[(size_t)d * (H * F) + f], alpha * hs_l[(size_t)s * (H * F) + f]);
}

__global__ void k_bias2(const float* __restrict__ agg, const float* __restrict__ b1,
                        const float* __restrict__ b2, float* __restrict__ out,
                        int Nn, int HF)
{
  int t = blockIdx.x * blockDim.x + threadIdx.x;
  if (t >= Nn * HF) return;
  int f = t % HF;
  float v = agg[t] + b1[f];
  if (b2) v += b2[f];
  out[t] = v;
}

__global__ void k_out_linear(const float* __restrict__ hp, const float* __restrict__ Wo,
                             const float* __restrict__ bo, float* __restrict__ out)
{
  int t = blockIdx.x * blockDim.x + threadIdx.x;
  if (t >= N_P) return;
  const float* __restrict__ row = hp + (size_t)t * 512;
  float s = bo[0];
  for (int k = 0; k < 512; ++k) s += row[k] * Wo[k];
  out[t] = s;
}

// ---------------------------------------------------------------------------
// Host-side orchestration
// ---------------------------------------------------------------------------
static inline void gemm(hipStream_t st, const float* A, const float* W, const float* bias,
                        float* C, int M, int N, int K, int ldc, int coff, int act)
{
  dim3 g((N + 15) / 16, (M + 15) / 16);
  k_gemm_wmma<<<g, 32, 0, st>>>(A, W, bias, C, M, N, K, ldc, coff, act);
}

struct GatScratch {
  float *hs_l, *hd_l, *esum, *a, *m_f, *z, *ssum, *dsum, *w_att, *mean, *loop_att;
  unsigned* m_ord;
};

static void run_gat(hipStream_t st,
                    const float* hs, int Ns, const float* hd, int Nd,
                    const int* src, const int* dst, int E, const float* efeat,
                    const float* Ws, const float* Wd,
                    const float* as_, const float* ad_,
                    const float* We, const float* ae_,
                    int H, int F, int Kin, bool self_loops,
                    float* agg, const GatScratch& S)
{
  const int HF = H * F;
  // linear projections (WMMA GEMMs)
  gemm(st, hs, Ws, nullptr, S.hs_l, Ns, HF, Kin, HF, 0, 0);
  const float* hd_l = S.hs_l;
  if (!(Ws == Wd && hs == hd)) {
    gemm(st, hd, Wd, nullptr, S.hd_l, Nd, HF, Kin, HF, 0, 0);
    hd_l = S.hd_l;
  }
  // edge-attention vector folding + per-edge/node attention scalars
  k_watt<<<(H * HID + 255) / 256, 256, 0, st>>>(ae_, We, S.w_att, H, F);
  k_esum<<<(E * H + 255) / 256, 256, 0, st>>>(efeat, S.w_att, S.esum, E, H);
  k_nodesum<<<(Ns * H + 255) / 256, 256, 0, st>>>(S.hs_l, as_, S.ssum, Ns, H, F);
  k_nodesum<<<(Nd * H + 255) / 256, 256, 0, st>>>(hd_l, ad_, S.dsum, Nd, H, F);
  int n_loop = self_loops ? Nd : 0;
  if (self_loops) {
    k_colmean<<<1, HID, 0, st>>>(efeat, S.mean, E);
    k_loop_att<<<1, 32, 0, st>>>(S.mean, S.w_att, S.loop_att, H);
  }
  int Etot = E + n_loop;
  // segment softmax
  k_fill_u32<<<(Nd * H + 255) / 256, 256, 0, st>>>(S.m_ord, ORD_NEG_INF, Nd * H);
  hipMemsetAsync(S.z, 0, (size_t)Nd * H * sizeof(float), st);
  k_attn<<<(Etot * H + 255) / 256, 256, 0, st>>>(S.ssum, S.dsum, S.esum, S.loop_att,
                                                 src, dst, E, n_loop, H, S.a, S.m_ord);
  k_m_decode<<<(Nd * H + 255) / 256, 256, 0, st>>>(S.m_ord, S.m_f, Nd * H);
  k_ea<<<(Etot * H + 255) / 256, 256, 0, st>>>(S.a, S.m_f, dst, E, n_loop, H, S.z);
  // scatter messages (accumulates into agg; caller zeroes per relation group)
  k_message<<<Etot, HF, 0, st>>>(S.a, S.z, S.hs_l, src, dst, E, H, F, agg);
}

extern "C" void kernel_launch(void* const* d_in, const int* in_sizes, int n_in,
                              void* d_out, int out_size, void* d_ws, size_t ws_size,
                              hipStream_t stream)
{
  (void)in_sizes; (void)n_in; (void)out_size; (void)ws_size;

  const float* x_proposal = (const float*)d_in[0];
  const float* x_branch   = (const float*)d_in[1];
  const float* x_patch    = (const float*)d_in[2];
  const int*   ei_pp      = (const int*)d_in[3];   // [2, E_PP]
  const int*   src_bp     = (const int*)d_in[4];
  const int*   dst_bp     = (const int*)d_in[5];
  const int*   ei_bb      = (const int*)d_in[6];   // [2, E_BB]
  const float* ea_pp      = (const float*)d_in[7];
  const float* ea_bp      = (const float*)d_in[8];
  const float* ea_bb      = (const float*)d_in[9];

  // params flattened in setup_inputs()/dict insertion order
  int p = 10;
  auto P = [&](void) { return (const float*)d_in[p++]; };
  const float *np_W1 = P(), *np_b1 = P(), *np_W2 = P(), *np_b2 = P();
  const float *nb_W1 = P(), *nb_b1 = P(), *nb_W2 = P(), *nb_b2 = P();
  const float *eppW1 = P(), *eppb1 = P(), *eppW2 = P(), *eppb2 = P();
  const float *ebpW1 = P(), *ebpb1 = P(), *ebpW2 = P(), *ebpb2 = P();
  const float *ebbW1 = P(), *ebbb1 = P(), *ebbW2 = P(), *ebbb2 = P();
  const float *pc_w1 = P(), *pc_b1 = P(), *pc_w2 = P(), *pc_b2 = P(), *pc_Wl = P(), *pc_bl = P();
  // gat_same: W, as, ad, We, ae, b  |  gat_mix: Ws, Wd, as, ad, We, ae, b
  const float *g1pp_W = P(), *g1pp_as = P(), *g1pp_ad = P(), *g1pp_We = P(), *g1pp_ae = P(), *g1pp_b = P();
  const float *g1bp_Ws = P(), *g1bp_Wd = P(), *g1bp_as = P(), *g1bp_ad = P(), *g1bp_We = P(), *g1bp_ae = P(), *g1bp_b = P();
  const float *g1bb_W = P(), *g1bb_as = P(), *g1bb_ad = P(), *g1bb_We = P(), *g1bb_ae = P(), *g1bb_b = P();
  const float *g2pp_W = P(), *g2pp_as = P(), *g2pp_ad = P(), *g2pp_We = P(), *g2pp_ae = P(), *g2pp_b = P();
  const float *g2bp_Ws = P(), *g2bp_Wd = P(), *g2bp_as = P(), *g2bp_ad = P(), *g2bp_We = P(), *g2bp_ae = P(), *g2bp_b = P();
  const float *g2bb_W = P(), *g2bb_as = P(), *g2bb_ad = P(), *g2bb_We = P(), *g2bb_ae = P(), *g2bb_b = P();
  const float *out_W = P(), *out_b = P();
  (void)g2bb_W; (void)g2bb_as; (void)g2bb_ad; (void)g2bb_We; (void)g2bb_ae; (void)g2bb_b; // o_bb layer2 unused

  // ---------------- workspace layout (floats) ----------------
  float* w = (float*)d_ws;
  size_t off = 0;
  auto F32 = [&](size_t n) { float* r = w + off; off += n; return r; };
  float* mlp_hidden = F32((size_t)E_BB * 256);         // shared MLP hidden
  float* epp        = F32((size_t)E_PP * HID);
  float* ebp        = F32((size_t)E_BP * HID);
  float* ebb        = F32((size_t)E_BB * HID);
  float* hp1        = F32((size_t)N_P * HID);          // [mlp(64) | patch(64)]
  float* hb1        = F32((size_t)N_B * HID);
  float* pooled     = F32((size_t)N_P * 16);
  float* conv1_buf  = F32((size_t)CHUNK * 8 * 25 * 25 * 25);
  float* conv2_buf  = F32((size_t)CHUNK * 16 * 13 * 13 * 13);
  GatScratch S;
  S.hs_l     = F32((size_t)N_B * 512);
  S.hd_l     = F32((size_t)N_P * 512);
  S.esum     = F32((size_t)E_BB * 2);
  S.a        = F32((size_t)(E_BB + N_B) * 2);
  S.m_f      = F32((size_t)N_B * 2);
  S.z        = F32((size_t)N_B * 2);
  S.ssum     = F32((size_t)N_B * 2);
  S.dsum     = F32((size_t)N_B * 2);
  S.w_att    = F32(2 * HID);
  S.mean     = F32(HID);
  S.loop_att = F32(2);
  S.m_ord    = (unsigned*)F32((size_t)N_B * 2);
  float* agg_p = F32((size_t)N_P * 512);
  float* agg_b = F32((size_t)N_B * 256);
  float* hp2   = F32((size_t)N_P * 256);
  float* hb2   = F32((size_t)N_B * 256);
  float* hp3   = F32((size_t)N_P * 512);

  // ---------------- patch embed (chunked over batch) ----------------
  for (int base = 0; base < N_P; base += CHUNK) {
    int n1 = CHUNK * 8 * 25 * 25 * 25;
    int n2 = CHUNK * 16 * 13 * 13 * 13;
    int n3 = CHUNK * 16;
    k_conv1<<<(n1 + 255) / 256, 256, 0, stream>>>(x_patch + (size_t)base * 125000,
                                                  pc_w1, pc_b1, conv1_buf, CHUNK);
    k_conv2<<<(n2 + 255) / 256, 256, 0, stream>>>(conv1_buf, pc_w2, pc_b2, conv2_buf, CHUNK);
    k_gap<<<(n3 + 255) / 256, 256, 0, stream>>>(conv2_buf, pooled + (size_t)base * 16, CHUNK);
  }
  k_patch_linear<<<(N_P * 64 + 255) / 256, 256, 0, stream>>>(pooled, pc_Wl, pc_bl, hp1);

  // ---------------- node / edge MLPs (WMMA GEMMs) ----------------
  gemm(stream, x_proposal, np_W1, np_b1, mlp_hidden, N_P, 128, D_P, 128, 0, 1);
  gemm(stream, mlp_hidden, np_W2, np_b2, hp1, N_P, 64, 128, HID, 0, 0); // cols 0..63
  gemm(stream, x_branch, nb_W1, nb_b1, mlp_hidden, N_B, 256, D_B, 256, 0, 1);
  gemm(stream, mlp_hidden, nb_W2, nb_b2, hb1, N_B, 128, 256, 128, 0, 0);
  gemm(stream, ea_pp, eppW1, eppb1, mlp_hidden, E_PP, 256, D_E, 256, 0, 1);
  gemm(stream, mlp_hidden, eppW2, eppb2, epp, E_PP, 128, 256, 128, 0, 0);
  gemm(stream, ea_bp, ebpW1, ebpb1, mlp_hidden, E_BP, 256, D_E, 256, 0, 1);
  gemm(stream, mlp_hidden, ebpW2, ebpb2, ebp, E_BP, 128, 256, 128, 0, 0);
  gemm(stream, ea_bb, ebbW1, ebbb1, mlp_hidden, E_BB, 256, D_E, 256, 0, 1);
  gemm(stream, mlp_hidden, ebbW2, ebbb2, ebb, E_BB, 128, 256, 128, 0, 0);

  // ---------------- HeteroConv layer 1 (heads=2, out=128) ----------------
  hipMemsetAsync(agg_p, 0, (size_t)N_P * 256 * sizeof(float), stream);
  hipMemsetAsync(agg_b, 0, (size_t)N_B * 256 * sizeof(float), stream);
  run_gat(stream, hp1, N_P, hp1, N_P, ei_pp, ei_pp + E_PP, E_PP, epp,
          g1pp_W, g1pp_W, g1pp_as, g1pp_ad, g1pp_We, g1pp_ae, 2, 128, 128, true, agg_p, S);
  run_gat(stream, hb1, N_B, hp1, N_P, src_bp, dst_bp, E_BP, ebp,
          g1bp_Ws, g1bp_Wd, g1bp_as, g1bp_ad, g1bp_We, g1bp_ae, 2, 128, 128, false, agg_p, S);
  run_gat(stream, hb1, N_B, hb1, N_B, ei_bb, ei_bb + E_BB, E_BB, ebb,
          g1bb_W, g1bb_W, g1bb_as, g1bb_ad, g1bb_We, g1bb_ae, 2, 128, 128, true, agg_b, S);
  k_bias2<<<((N_P * 256) + 255) / 256, 256, 0, stream>>>(agg_p, g1pp_b, g1bp_b, hp2, N_P, 256);
  k_bias2<<<((N_B * 256) + 255) / 256, 256, 0, stream>>>(agg_b, g1bb_b, nullptr, hb2, N_B, 256);

  // ---------------- HeteroConv layer 2 (heads=2, out=256; o_bb unused) ----------------
  hipMemsetAsync(agg_p, 0, (size_t)N_P * 512 * sizeof(float), stream);
  run_gat(stream, hp2, N_P, hp2, N_P, ei_pp, ei_pp + E_PP, E_PP, epp,
          g2pp_W, g2pp_W, g2pp_as, g2pp_ad, g2pp_We, g2pp_ae, 2, 256, 256, true, agg_p, S);
  run_gat(stream, hb2, N_B, hp2, N_P, src_bp, dst_bp, E_BP, ebp,
          g2bp_Ws, g2bp_Wd, g2bp_as, g2bp_ad, g2bp_We, g2bp_ae, 2, 256, 256, false, agg_p, S);
  k_bias2<<<((N_P * 512) + 255) / 256, 256, 0, stream>>>(agg_p, g2pp_b, g2bp_b, hp3, N_P, 512);

  // ---------------- output head ----------------
  k_out_linear<<<(N_P + 255) / 256, 256, 0, stream>>>(hp3, out_W, out_b, (float*)d_out);
}